// CloudCrop_multi_scale_33612414058692
// MI455X (gfx1250) — compile-verified
//
#include <hip/hip_runtime.h>
#include <math.h>

typedef float v2f __attribute__((ext_vector_type(2)));
typedef float v8f __attribute__((ext_vector_type(8)));

#define BQ     2
#define MQ     1024
#define CSEED  512
#define C1Q    515           // 3 + 512
#define C1PAD  516           // padded to multiple of 4 (row 515 zeroed)
#define NSAMP  16
#define NSC    4
#define HMINQ  (-0.02f)
#define HMAXQ  (0.04f)
#define EPSQ   (1e-5f)

__device__ __constant__ float kRadii[4] = {0.0125f, 0.025f, 0.0375f, 0.05f};

// ---------------------------------------------------------------------------
// Kernel A: per (b, m, scale) pick the first 16 neighbors (ascending n) that
// pass the rotated cylinder test; emit indices + g_xyz = local/radius.
// Block = 128 threads = 4 waves (one wave per scale), 8 centers per block.
// ---------------------------------------------------------------------------
__global__ __launch_bounds__(128) void select_kernel(
    const float* __restrict__ xyz, const float* __restrict__ rotm,
    int* __restrict__ ws_idx, float* __restrict__ ws_gxyz)
{
  __shared__ float sx[MQ], sy[MQ], sz[MQ];
  __shared__ int   sidx[4][NSAMP];
  __shared__ float sgx[4][NSAMP], sgy[4][NSAMP], sgz[4][NSAMP];

  const int tid  = threadIdx.x;
  const int b    = blockIdx.x >> 7;          // MQ/8 = 128 groups per batch
  const int mg   = blockIdx.x & 127;
  const int wv   = __builtin_amdgcn_readfirstlane(tid >> 5);   // scale id (scalar)
  const int lane = tid & 31;

  for (int i = tid; i < MQ; i += 128) {
    const float* p = xyz + ((size_t)b * MQ + i) * 3;
    sx[i] = p[0]; sy[i] = p[1]; sz[i] = p[2];
  }
  __syncthreads();

  const float radius = kRadii[wv];
  const float r2     = radius * radius;
  const float inv_r  = 1.0f / radius;

  for (int mi = 0; mi < 8; ++mi) {
    const int m = mg * 8 + mi;
    const float* R = rotm + ((size_t)b * MQ + m) * 9;
    const float r00=R[0], r01=R[1], r02=R[2];
    const float r10=R[3], r11=R[4], r12=R[5];
    const float r20=R[6], r21=R[7], r22=R[8];
    const float cx = sx[m], cy = sy[m], cz = sz[m];

    int count = 0;
    for (int base = 0; base < MQ && count < NSAMP; base += 32) {
      const int n = base + lane;
      const float dx = sx[n]-cx, dy = sy[n]-cy, dz = sz[n]-cz;
      const float lx = dx*r00 + dy*r10 + dz*r20;
      const float ly = dx*r01 + dy*r11 + dz*r21;
      const float lz = dx*r02 + dy*r12 + dz*r22;
      const bool ok = (ly*ly + lz*lz < r2) && (lx > HMINQ) && (lx < HMAXQ);
      const unsigned bm = (unsigned)__ballot(ok);           // wave32: low 32 bits
      const int pos = count + __popc(bm & ((1u << lane) - 1u));
      if (ok && pos < NSAMP) {
        sidx[wv][pos] = n;
        sgx[wv][pos] = lx * inv_r;
        sgy[wv][pos] = ly * inv_r;
        sgz[wv][pos] = lz * inv_r;
      }
      count += __popc(bm);
    }
    int nvalid = (count < NSAMP) ? count : NSAMP;
    if (nvalid == 0) {
      // argsort(all-ones) -> idx == 0 everywhere
      const float dx = sx[0]-cx, dy = sy[0]-cy, dz = sz[0]-cz;
      const float lx = dx*r00 + dy*r10 + dz*r20;
      const float ly = dx*r01 + dy*r11 + dz*r21;
      const float lz = dx*r02 + dy*r12 + dz*r22;
      if (lane == 0) {
        sidx[wv][0] = 0;
        sgx[wv][0] = lx * inv_r; sgy[wv][0] = ly * inv_r; sgz[wv][0] = lz * inv_r;
      }
      nvalid = 1;
    }
    if (lane < NSAMP) {
      const int src = (lane < nvalid) ? lane : 0;           // pad with slot 0
      const size_t o = ((size_t)(b * NSC + wv)) * MQ + m;
      ws_idx[o * NSAMP + lane] = sidx[wv][src];
      ws_gxyz[o * 48 + 0 * 16 + lane] = sgx[wv][src];
      ws_gxyz[o * 48 + 1 * 16 + lane] = sgy[wv][src];
      ws_gxyz[o * 48 + 2 * 16 + lane] = sgz[wv][src];
    }
  }
}

// ---------------------------------------------------------------------------
// Kernel B: per (b, m, s): X(516x16) in LDS; conv1 (256x515) + BN+ReLU,
// conv2 (256x256) + BN+ReLU, max over 16 samples -> cat[b, s*256+ch, m].
// 8 waves; each wave owns output row-tiles {wv, wv+8}; the tile loop has a
// compile-time trip count (2) so no EXEC-masked control flow wraps the WMMAs.
// ---------------------------------------------------------------------------
__global__ __launch_bounds__(256) void crop_kernel(
    const float* __restrict__ feats,
    const float* __restrict__ W1, const float* __restrict__ b1,
    const float* __restrict__ g1, const float* __restrict__ be1,
    const float* __restrict__ m1, const float* __restrict__ v1,
    const float* __restrict__ W2, const float* __restrict__ b2,
    const float* __restrict__ g2, const float* __restrict__ be2,
    const float* __restrict__ m2, const float* __restrict__ v2,
    const int* __restrict__ ws_idx, const float* __restrict__ ws_gxyz,
    float* __restrict__ cat)
{
  __shared__ float Xl [C1PAD * NSAMP];   // 516 x 16 (K x N)
  __shared__ float X2l[256 * NSAMP];     // 256 x 16
  __shared__ float sc1[256], sh1[256], sc2[256], sh2[256];
  __shared__ int   idxl[NSAMP];

  const int tid = threadIdx.x;
  const size_t job = blockIdx.x;         // ((b*4)+s)*MQ + m
  const int m  = (int)(job & (MQ - 1));
  const int bs = (int)(job >> 10);
  const int s  = bs & 3;
  const int b  = bs >> 2;

  if (tid < NSAMP)                 idxl[tid] = ws_idx[job * NSAMP + tid];
  if (tid < 48)                    Xl[tid]   = ws_gxyz[job * 48 + tid];   // rows 0..2
  if (tid >= 48 && tid < 64)       Xl[C1Q * 16 + (tid - 48)] = 0.0f;      // zero pad row
  {
    const int ch = tid;                     // 256 threads == 256 channels
    const float s1 = g1[s*256+ch] * rsqrtf(v1[s*256+ch] + EPSQ);
    sc1[ch] = s1;
    sh1[ch] = (b1[s*256+ch] - m1[s*256+ch]) * s1 + be1[s*256+ch];
    const float s2 = g2[s*256+ch] * rsqrtf(v2[s*256+ch] + EPSQ);
    sc2[ch] = s2;
    sh2[ch] = (b2[s*256+ch] - m2[s*256+ch]) * s2 + be2[s*256+ch];
  }
  __syncthreads();                          // idxl visible

  for (int i = tid; i < CSEED * NSAMP; i += 256) {
    const int c = i >> 4, n = i & 15;
    Xl[(3 + c) * 16 + n] = feats[((size_t)b * CSEED + c) * MQ + idxl[n]];
  }
  __syncthreads();

  const int wv   = __builtin_amdgcn_readfirstlane(tid >> 5);   // wave id (scalar)
  const int lane = tid & 31;
  const int half = lane >> 4, col = lane & 15;

  // ---- conv1: 256x515 @ 515x16 ----
  const float* W1s = W1 + (size_t)s * 256 * C1Q;
  #pragma unroll
  for (int tt = 0; tt < 2; ++tt) {
    const int t = wv + 8 * tt;
    v8f acc = {0.f,0.f,0.f,0.f,0.f,0.f,0.f,0.f};
    const float* Wr = W1s + (size_t)(16 * t + col) * C1Q;  // A row = out channel
    __builtin_prefetch(Wr, 0, 1);
    #pragma unroll 4
    for (int k0 = 0; k0 < C1PAD; k0 += 4) {
      const int ka = k0 + 2 * half;
      v2f a, bf;
      a.x  = (ka     < C1Q) ? Wr[ka]     : 0.0f;
      a.y  = (ka + 1 < C1Q) ? Wr[ka + 1] : 0.0f;
      bf.x = Xl[ka * 16 + col];
      bf.y = Xl[(ka + 1) * 16 + col];
      acc = __builtin_amdgcn_wmma_f32_16x16x4_f32(false, a, false, bf,
                                                  (short)0, acc, false, false);
    }
    #pragma unroll
    for (int r = 0; r < 8; ++r) {
      const int ch = 16 * t + r + 8 * half;
      X2l[ch * 16 + col] = fmaxf(acc[r] * sc1[ch] + sh1[ch], 0.0f);
    }
  }
  __syncthreads();

  // ---- conv2: 256x256 @ 256x16, then row-max over 16 samples ----
  const float* W2s = W2 + (size_t)s * 256 * 256;
  #pragma unroll
  for (int tt = 0; tt < 2; ++tt) {
    const int t = wv + 8 * tt;
    v8f acc = {0.f,0.f,0.f,0.f,0.f,0.f,0.f,0.f};
    const float* Wr = W2s + (size_t)(16 * t + col) * 256;
    __builtin_prefetch(Wr, 0, 1);
    #pragma unroll 4
    for (int k0 = 0; k0 < 256; k0 += 4) {
      const int ka = k0 + 2 * half;
      v2f a, bf;
      a.x  = Wr[ka];
      a.y  = Wr[ka + 1];
      bf.x = X2l[ka * 16 + col];
      bf.y = X2l[(ka + 1) * 16 + col];
      acc = __builtin_amdgcn_wmma_f32_16x16x4_f32(false, a, false, bf,
                                                  (short)0, acc, false, false);
    }
    #pragma unroll
    for (int r = 0; r < 8; ++r) {
      const int ch = 16 * t + r + 8 * half;
      float y = fmaxf(acc[r] * sc2[ch] + sh2[ch], 0.0f);
      #pragma unroll
      for (int off = 1; off < 16; off <<= 1)          // max within 16-lane group
        y = fmaxf(y, __shfl_xor(y, off, 32));
      if (col == 0)
        cat[((size_t)b * 1024 + s * 256 + ch) * MQ + m] = y;
    }
  }
}

// ---------------------------------------------------------------------------
// Kernel C: per (b, 16-column m tile): seed_t = trans_W@feats + trans_b (LDS),
// out = fuse_W@cat + fuse_b + sigmoid(gate_W@seed_t + gate_b) * seed_t.
// ---------------------------------------------------------------------------
__global__ __launch_bounds__(256) void fuse_kernel(
    const float* __restrict__ feats,
    const float* __restrict__ fuseW, const float* __restrict__ fuseB,
    const float* __restrict__ gateW, const float* __restrict__ gateB,
    const float* __restrict__ transW, const float* __restrict__ transB,
    const float* __restrict__ cat, float* __restrict__ out)
{
  __shared__ float stl[256 * NSAMP];     // seed_t tile, 256 x 16

  const int tid = threadIdx.x;
  const int b  = blockIdx.x >> 6;
  const int m0 = (blockIdx.x & 63) * 16;
  const int wv   = __builtin_amdgcn_readfirstlane(tid >> 5);
  const int lane = tid & 31;
  const int half = lane >> 4, col = lane & 15;

  // ---- seed_t: 256x512 @ 512x16 ----
  #pragma unroll
  for (int tt = 0; tt < 2; ++tt) {
    const int t = wv + 8 * tt;
    v8f acc = {0.f,0.f,0.f,0.f,0.f,0.f,0.f,0.f};
    const float* Wr = transW + (size_t)(16 * t + col) * CSEED;
    const float* F  = feats + (size_t)b * CSEED * MQ + m0 + col;
    __builtin_prefetch(Wr, 0, 1);
    #pragma unroll 4
    for (int k0 = 0; k0 < CSEED; k0 += 4) {
      const int ka = k0 + 2 * half;
      v2f a, bf;
      a.x  = Wr[ka];
      a.y  = Wr[ka + 1];
      bf.x = F[(size_t)ka * MQ];
      bf.y = F[(size_t)(ka + 1) * MQ];
      acc = __builtin_amdgcn_wmma_f32_16x16x4_f32(false, a, false, bf,
                                                  (short)0, acc, false, false);
    }
    #pragma unroll
    for (int r = 0; r < 8; ++r) {
      const int ch = 16 * t + r + 8 * half;
      stl[ch * 16 + col] = acc[r] + transB[ch];
    }
  }
  __syncthreads();

  // ---- fused (K=1024 from cat) + gate (K=256 from stl) ----
  const float* Cb = cat + (size_t)b * 1024 * MQ + m0 + col;
  #pragma unroll
  for (int tt = 0; tt < 2; ++tt) {
    const int t = wv + 8 * tt;
    v8f accF = {0.f,0.f,0.f,0.f,0.f,0.f,0.f,0.f};
    v8f accG = {0.f,0.f,0.f,0.f,0.f,0.f,0.f,0.f};
    const float* WrF = fuseW + (size_t)(16 * t + col) * 1024;
    const float* WrG = gateW + (size_t)(16 * t + col) * 256;
    __builtin_prefetch(WrF, 0, 1);
    #pragma unroll 4
    for (int k0 = 0; k0 < 1024; k0 += 4) {
      const int ka = k0 + 2 * half;
      v2f a, bf;
      a.x  = WrF[ka];
      a.y  = WrF[ka + 1];
      bf.x = Cb[(size_t)ka * MQ];
      bf.y = Cb[(size_t)(ka + 1) * MQ];
      accF = __builtin_amdgcn_wmma_f32_16x16x4_f32(false, a, false, bf,
                                                   (short)0, accF, false, false);
    }
    #pragma unroll 4
    for (int k0 = 0; k0 < 256; k0 += 4) {
      const int ka = k0 + 2 * half;
      v2f a, bf;
      a.x  = WrG[ka];
      a.y  = WrG[ka + 1];
      bf.x = stl[ka * 16 + col];
      bf.y = stl[(ka + 1) * 16 + col];
      accG = __builtin_amdgcn_wmma_f32_16x16x4_f32(false, a, false, bf,
                                                   (short)0, accG, false, false);
    }
    #pragma unroll
    for (int r = 0; r < 8; ++r) {
      const int ch = 16 * t + r + 8 * half;
      const float st = stl[ch * 16 + col];
      const float gl = accG[r] + gateB[ch];
      const float gate = 1.0f / (1.0f + expf(-gl));
      out[((size_t)b * 256 + ch) * MQ + m0 + col] = accF[r] + fuseB[ch] + gate * st;
    }
  }
}

// ---------------------------------------------------------------------------
extern "C" void kernel_launch(void* const* d_in, const int* in_sizes, int n_in,
                              void* d_out, int out_size, void* d_ws, size_t ws_size,
                              hipStream_t stream) {
  const float* xyz    = (const float*)d_in[0];
  const float* feats  = (const float*)d_in[1];
  const float* rot    = (const float*)d_in[2];
  const float* W1     = (const float*)d_in[3];
  const float* b1     = (const float*)d_in[4];
  const float* g1     = (const float*)d_in[5];
  const float* be1    = (const float*)d_in[6];
  const float* m1     = (const float*)d_in[7];
  const float* v1     = (const float*)d_in[8];
  const float* W2     = (const float*)d_in[9];
  const float* b2     = (const float*)d_in[10];
  const float* g2     = (const float*)d_in[11];
  const float* be2    = (const float*)d_in[12];
  const float* m2     = (const float*)d_in[13];
  const float* v2     = (const float*)d_in[14];
  const float* fuseW  = (const float*)d_in[15];
  const float* fuseB  = (const float*)d_in[16];
  const float* gateW  = (const float*)d_in[17];
  const float* gateB  = (const float*)d_in[18];
  const float* transW = (const float*)d_in[19];
  const float* transB = (const float*)d_in[20];

  char* ws = (char*)d_ws;
  int*   ws_idx  = (int*)ws;                          //  524,288 B
  float* ws_gxyz = (float*)(ws + 524288);             // 1,572,864 B
  float* catbuf  = (float*)(ws + 2097152);            // 8,388,608 B

  select_kernel<<<BQ * (MQ / 8), 128, 0, stream>>>(xyz, rot, ws_idx, ws_gxyz);
  crop_kernel<<<BQ * NSC * MQ, 256, 0, stream>>>(feats, W1, b1, g1, be1, m1, v1,
                                                 W2, b2, g2, be2, m2, v2,
                                                 ws_idx, ws_gxyz, catbuf);
  fuse_kernel<<<BQ * (MQ / 16), 256, 0, stream>>>(feats, fuseW, fuseB, gateW, gateB,
                                                  transW, transB, catbuf,
                                                  (float*)d_out);
}